// HexagonalSensor_57475252355472
// MI455X (gfx1250) — compile-verified
//
#include <hip/hip_runtime.h>
#include <math.h>

#define BLOCK 256
#define MAX_BLOCKS 1024

__global__ void hex_zero_f32(float* __restrict__ p, int n) {
    int i = blockIdx.x * blockDim.x + threadIdx.x;
    if (i < n) p[i] = 0.0f;
}

__device__ __forceinline__ void hex_bin_one(
    float px, float py, float pv,
    float ox, float oy, float ca, float sa, float inv_hs,
    int qmin, int rmin, int Q, int R,
    const int* __restrict__ lut, float* __restrict__ hist)
{
    const float kq       = 0.57735026918962576451f; // sqrt(3)/3
    const float third    = 1.0f / 3.0f;
    const float twothird = 2.0f / 3.0f;

    float xs = px - ox;
    float ys = py - oy;
    float xr = ca * xs - sa * ys;
    float yr = sa * xs + ca * ys;
    float qf = (kq * xr - yr * third) * inv_hs;
    float rf = (twothird * yr) * inv_hs;
    float sf = -qf - rf;
    // round-half-even matches jnp.round -> v_rndne_f32
    float qr = rintf(qf), rr = rintf(rf), sr = rintf(sf);
    float qd = fabsf(qr - qf), rd = fabsf(rr - rf), sd = fabsf(sr - sf);
    // conditions are mutually exclusive; reference applies them sequentially
    if ((qd > rd) && (qd > sd))      qr = -rr - sr;
    else if ((rd > qd) && (rd > sd)) rr = -qr - sr;

    int qi = (int)qr - qmin;
    int ri = (int)rr - rmin;
    if ((unsigned)qi < (unsigned)Q && (unsigned)ri < (unsigned)R) {
        int pix = lut[qi * R + ri];
        if (pix >= 0) {
            // LDS float atomic -> ds_add_f32 (no return), confirmed in round-1 disasm
            __hip_atomic_fetch_add(&hist[pix], pv,
                                   __ATOMIC_RELAXED, __HIP_MEMORY_SCOPE_WORKGROUP);
        }
    }
}

__global__ __launch_bounds__(BLOCK) void hex_hist_kernel(
    const float* __restrict__ xp, const float* __restrict__ yp,
    const float* __restrict__ vp, const int* __restrict__ lut,
    const float* __restrict__ s_hex, const float* __restrict__ s_rot,
    const float* __restrict__ s_ox,  const float* __restrict__ s_oy,
    const int* __restrict__ s_qmin,  const int* __restrict__ s_rmin,
    int Q, int R, int npix, int n, float* __restrict__ out)
{
    extern __shared__ float hist[];   // npix floats (7.2 KB of 320 KB/WGP)
    for (int i = threadIdx.x; i < npix; i += BLOCK) hist[i] = 0.0f;
    __syncthreads();

    // uniform scalars (scalar loads)
    const float hs  = s_hex[0];
    const float rot = s_rot[0];
    const float ox  = s_ox[0],  oy  = s_oy[0];
    const int  qmin = s_qmin[0], rmin = s_rmin[0];
    // uniform rotation; hardware v_cos/v_sin are exact at 0 and avoid the
    // Payne-Hanek slow path (big SALU + rodata loads seen in round 1)
    const float ca = __cosf(-rot);
    const float sa = __sinf(-rot);
    const float inv_hs = 1.0f / hs;

    const int n4 = n >> 2;
    const float4* x4 = (const float4*)xp;
    const float4* y4 = (const float4*)yp;
    const float4* v4 = (const float4*)vp;
    const int stride = gridDim.x * BLOCK;

    for (int i = blockIdx.x * BLOCK + threadIdx.x; i < n4; i += stride) {
        // pull the next grid-stride tile toward L2 (global_prefetch_b8,
        // speculative: OOB translations are silently dropped)
        __builtin_prefetch(x4 + i + stride, 0, 0);
        __builtin_prefetch(y4 + i + stride, 0, 0);
        __builtin_prefetch(v4 + i + stride, 0, 0);

        float4 X = x4[i];   // global_load_b128
        float4 Y = y4[i];
        float4 V = v4[i];
        hex_bin_one(X.x, Y.x, V.x, ox, oy, ca, sa, inv_hs, qmin, rmin, Q, R, lut, hist);
        hex_bin_one(X.y, Y.y, V.y, ox, oy, ca, sa, inv_hs, qmin, rmin, Q, R, lut, hist);
        hex_bin_one(X.z, Y.z, V.z, ox, oy, ca, sa, inv_hs, qmin, rmin, Q, R, lut, hist);
        hex_bin_one(X.w, Y.w, V.w, ox, oy, ca, sa, inv_hs, qmin, rmin, Q, R, lut, hist);
    }
    // tail (n not divisible by 4)
    for (int i = (n4 << 2) + blockIdx.x * BLOCK + threadIdx.x; i < n; i += stride) {
        hex_bin_one(xp[i], yp[i], vp[i], ox, oy, ca, sa, inv_hs,
                    qmin, rmin, Q, R, lut, hist);
    }

    __syncthreads();
    // Staggered flush: different blocks start at different bins so the
    // per-address L2 atomic serialization is spread across channels.
    int start = (int)((blockIdx.x * 61u) % (unsigned)npix);
    for (int t = threadIdx.x; t < npix; t += BLOCK) {
        int i = start + t;
        if (i >= npix) i -= npix;
        float h = hist[i];
        if (h != 0.0f) {
            // force hardware global_atomic_add_f32 (no CAS retry loop)
            unsafeAtomicAdd(&out[i], h);
        }
    }
}

extern "C" void kernel_launch(void* const* d_in, const int* in_sizes, int n_in,
                              void* d_out, int out_size, void* d_ws, size_t ws_size,
                              hipStream_t stream) {
    (void)n_in; (void)d_ws; (void)ws_size;
    const float* xp   = (const float*)d_in[0];
    const float* yp   = (const float*)d_in[1];
    const float* vp   = (const float*)d_in[2];
    const int*   lut  = (const int*)d_in[3];
    const float* hs   = (const float*)d_in[4];
    const float* rot  = (const float*)d_in[5];
    const float* ox   = (const float*)d_in[6];
    const float* oy   = (const float*)d_in[7];
    const int*   qmin = (const int*)d_in[8];
    const int*   rmin = (const int*)d_in[9];
    float* out = (float*)d_out;

    const int n    = in_sizes[0];
    const int npix = out_size;

    // lookup table is square (2R_grid+1)^2; recover side length host-side
    const int lutN = in_sizes[3];
    int dim = 1;
    while ((long long)dim * dim < (long long)lutN) dim++;
    const int Q = dim;
    const int R = lutN / dim;

    // d_out is poisoned by the harness and not re-zeroed between replays
    hex_zero_f32<<<(npix + BLOCK - 1) / BLOCK, BLOCK, 0, stream>>>(out, npix);

    int n4 = n >> 2;
    int nblocks = (n4 + BLOCK - 1) / BLOCK;
    if (nblocks > MAX_BLOCKS) nblocks = MAX_BLOCKS;
    if (nblocks < 1) nblocks = 1;

    hex_hist_kernel<<<nblocks, BLOCK, (size_t)npix * sizeof(float), stream>>>(
        xp, yp, vp, lut, hs, rot, ox, oy, qmin, rmin, Q, R, npix, n, out);
}